// GliomaPhysicsModel_62448824484443
// MI455X (gfx1250) — compile-verified
//
#include <hip/hip_runtime.h>
#include <hip/hip_bf16.h>

// ---------------------------------------------------------------------------
// Glioma physics model on MI455X (gfx1250).
// conv1 (4->16, 3^3) and conv2 (16->32, 3^3) run as implicit GEMM on the fp32
// WMMA path: V_WMMA_F32_16X16X4_F32, M=out-channel tile(16), N=16 x-positions
// per wave, K = C_in chunked by 4, accumulated over the 27 spatial taps.
// Fully unrolled inner loops: per K-chunk = 1 ds_load_b64 (B, interleaved LDS)
// + 1-2 global_load_b64 (A, immediate offsets) + 1-2 v_wmma.
// ---------------------------------------------------------------------------

typedef __attribute__((ext_vector_type(2))) float v2f;
typedef __attribute__((ext_vector_type(8))) float v8f;

static constexpr int    DIM  = 128;
static constexpr size_t VOL  = (size_t)DIM * DIM * DIM;   // 2,097,152
static constexpr int    ROWL = 66;                        // 64 + 2 halo

// ---------------------------------------------------------------------------
// Weight re-layout: w[oc][ci][tap] -> wt[((tap*(CIN/2) + ci/2)*COUT + oc)*2 + (ci&1)]
// so each lane's A fragment (two consecutive K values) is one contiguous v2f.
// ---------------------------------------------------------------------------
template <int CIN, int COUT>
__global__ void wprep_kernel(const float* __restrict__ w, float* __restrict__ wt) {
    int i = blockIdx.x * 256 + threadIdx.x;
    constexpr int TOTAL = COUT * CIN * 27;
    if (i >= TOTAL) return;
    int oc  = i / (CIN * 27);
    int r   = i - oc * (CIN * 27);
    int ci  = r / 27;
    int tap = r - ci * 27;
    wt[((tap * (CIN / 2) + (ci >> 1)) * COUT + oc) * 2 + (ci & 1)] = w[i];
}

__global__ void zero_kernel(float* p, int n) {
    int i = blockIdx.x * 64 + threadIdx.x;
    if (i < n) p[i] = 0.0f;
}

// ---------------------------------------------------------------------------
// Implicit-GEMM 3x3x3 convolution via fp32 WMMA.
// Grid: (2, 128, 128) = (x-block of 64, y, z). Block: 128 threads = 4 waves.
// Wave w handles x in [xb + 16w, xb + 16w + 16), all COUT channels.
// WMMA fragment layouts (CDNA5 ISA 7.12.2):
//   A 16x4 : lanes 0-15 -> M=lane, K={c0,c0+1}; lanes 16-31 -> K={c0+2,c0+3}
//   B 4x16 : lanes 0-15 -> N=lane, K={c0,c0+1}; lanes 16-31 -> K={c0+2,c0+3}
//   C 16x16: VGPR j -> M = j + 8*(lane>=16), N = lane&15
// LDS layout interleaved by channel pair: sIn[cpair][plane(9)][x(66)][2]
// so a B fragment is a single aligned 8-byte LDS load.
// ---------------------------------------------------------------------------
template <int CIN, int COUT>
__global__ __launch_bounds__(128) void conv3_wmma_kernel(
    const float* __restrict__ in, const float* __restrict__ wt,
    const float* __restrict__ bias, float* __restrict__ out) {

    __shared__ float sIn[CIN * 9 * ROWL];   // (CIN/2)*9*66 v2f elements

    const int tid = threadIdx.x;
    const int xb  = blockIdx.x * 64;
    const int yy  = blockIdx.y;
    const int zz  = blockIdx.z;

    // Cooperative halo load (zero padding at volume borders), interleaved store.
    for (int i = tid; i < CIN * 9 * ROWL; i += 128) {
        int xi = i % ROWL;
        int r  = i / ROWL;
        int dy = r % 3; r /= 3;
        int dz = r % 3;
        int c  = r / 3;
        int gx = xb - 1 + xi, gy = yy + dy - 1, gz = zz + dz - 1;
        float v = 0.0f;
        if (gx >= 0 && gx < DIM && gy >= 0 && gy < DIM && gz >= 0 && gz < DIM)
            v = in[(((size_t)c * DIM + gz) * DIM + gy) * DIM + gx];
        sIn[(((c >> 1) * 9 + (dz * 3 + dy)) * ROWL + xi) * 2 + (c & 1)] = v;
    }
    __syncthreads();

    const int lane = tid & 31;
    const int wv   = tid >> 5;
    const int n    = lane & 15;      // N index (x offset within wave tile)
    const int half = lane >> 4;      // K half selector
    const int xl   = wv * 16 + n;    // column into sIn rows (halo-based)

    v8f acc0, acc1;
#pragma unroll
    for (int j = 0; j < 8; ++j) {
        acc0[j] = bias[j + 8 * half];
        if constexpr (COUT == 32) acc1[j] = bias[16 + j + 8 * half];
        else                      acc1[j] = 0.0f;
    }

    // Per-lane base pointers; all further addressing is immediate offsets.
    const v2f* aBase = (const v2f*)wt + (size_t)half * COUT + n;
    const v2f* bBase = (const v2f*)sIn + (size_t)half * (9 * ROWL) + xl;

#pragma unroll
    for (int dz = 0; dz < 3; ++dz) {
#pragma unroll
        for (int dy = 0; dy < 3; ++dy) {
            const int plane = dz * 3 + dy;
#pragma unroll
            for (int dx = 0; dx < 3; ++dx) {
#pragma unroll
                for (int c0 = 0; c0 < CIN; c0 += 4) {
                    // chunk index: advances by 1 per (tap, c0) step
                    const int chunk = (plane * 3 + dx) * (CIN / 4) + (c0 >> 2);
                    // B fragment: cpair = c0/2 + half -> base already holds half
                    const v2f b = bBase[(size_t)(c0 >> 1) * (9 * ROWL) + plane * ROWL + dx];
                    const v2f a0 = aBase[(size_t)chunk * 2 * COUT];
                    acc0 = __builtin_amdgcn_wmma_f32_16x16x4_f32(
                        false, a0, false, b, (short)0, acc0, false, false);
                    if constexpr (COUT == 32) {
                        const v2f a1 = aBase[(size_t)chunk * 2 * COUT + 16];
                        acc1 = __builtin_amdgcn_wmma_f32_16x16x4_f32(
                            false, a1, false, b, (short)0, acc1, false, false);
                    }
                }
            }
        }
    }

    const size_t sp = ((size_t)zz * DIM + yy) * DIM + xb + wv * 16 + n;
#pragma unroll
    for (int j = 0; j < 8; ++j) {
        out[(size_t)(j + 8 * half) * VOL + sp] = acc0[j];
        if constexpr (COUT == 32)
            out[(size_t)(16 + j + 8 * half) * VOL + sp] = acc1[j];
    }
}

// ---------------------------------------------------------------------------
// GroupNorm statistics: both GNs have 4 channels/group -> group = ch >> 2.
// Grid: (VOL/8192, C). Block reduce then one atomicAdd pair per block.
// ---------------------------------------------------------------------------
__global__ void gn_stats_kernel(const float* __restrict__ y, float* __restrict__ stats) {
    const int ch = blockIdx.y;
    const int g  = ch >> 2;
    const float* p = y + (size_t)ch * VOL + (size_t)blockIdx.x * 8192;
    float s = 0.0f, q = 0.0f;
    for (int i = threadIdx.x; i < 8192; i += 256) {
        float v = p[i];
        s += v; q += v * v;
    }
    __shared__ float rs[256], rq[256];
    rs[threadIdx.x] = s; rq[threadIdx.x] = q;
    __syncthreads();
    for (int off = 128; off > 0; off >>= 1) {
        if (threadIdx.x < off) {
            rs[threadIdx.x] += rs[threadIdx.x + off];
            rq[threadIdx.x] += rq[threadIdx.x + off];
        }
        __syncthreads();
    }
    if (threadIdx.x == 0) {
        atomicAdd(&stats[g * 2 + 0], rs[0]);
        atomicAdd(&stats[g * 2 + 1], rq[0]);
    }
}

// GN + SiLU applied in place (layer 1).
__global__ void gn_silu_kernel(float* __restrict__ y, const float* __restrict__ stats,
                               const float* __restrict__ gam, const float* __restrict__ bet) {
    const size_t i  = (size_t)blockIdx.x * 256 + threadIdx.x;
    const int    ch = (int)(i >> 21);          // / VOL
    const int    g  = ch >> 2;
    const float  cnt = 4.0f * (float)VOL;
    const float  mu  = stats[g * 2] / cnt;
    const float  var = stats[g * 2 + 1] / cnt - mu * mu;
    float xn = (y[i] - mu) * rsqrtf(var + 1e-5f) * gam[ch] + bet[ch];
    y[i] = xn / (1.0f + expf(-xn));
}

// GN2 + SiLU fused with conv3 (1x1x1, 32->2) + sigmoid*0.5 -> D, rho.
__global__ void gn2_head_kernel(const float* __restrict__ y2, const float* __restrict__ stats,
                                const float* __restrict__ gam, const float* __restrict__ bet,
                                const float* __restrict__ w3, const float* __restrict__ b3,
                                float* __restrict__ Dout, float* __restrict__ Rout) {
    const size_t v = (size_t)blockIdx.x * 256 + threadIdx.x;
    const float cnt = 4.0f * (float)VOL;
    float aD = b3[0], aR = b3[1];
#pragma unroll
    for (int ch = 0; ch < 32; ++ch) {
        const int g = ch >> 2;
        const float mu  = stats[g * 2] / cnt;
        const float var = stats[g * 2 + 1] / cnt - mu * mu;
        float xn = (y2[(size_t)ch * VOL + v] - mu) * rsqrtf(var + 1e-5f) * gam[ch] + bet[ch];
        float h  = xn / (1.0f + expf(-xn));
        aD += w3[ch] * h;
        aR += w3[32 + ch] * h;
    }
    Dout[v] = 0.5f / (1.0f + expf(-aD));
    Rout[v] = 0.5f / (1.0f + expf(-aR));
}

// One reaction-diffusion step: 7-point Laplacian (zero padding) + logistic term.
__global__ void rd_step_kernel(const float* __restrict__ c, float* __restrict__ cn,
                               const float* __restrict__ Dv, const float* __restrict__ Rv,
                               const float* __restrict__ dtp) {
    const size_t i = (size_t)blockIdx.x * 256 + threadIdx.x;
    const int x = (int)(i & 127);
    const int y = (int)((i >> 7) & 127);
    const int z = (int)(i >> 14);
    const float cc = c[i];
    float lap = -6.0f * cc;
    lap += (x > 0)   ? c[i - 1]     : 0.0f;
    lap += (x < 127) ? c[i + 1]     : 0.0f;
    lap += (y > 0)   ? c[i - 128]   : 0.0f;
    lap += (y < 127) ? c[i + 128]   : 0.0f;
    lap += (z > 0)   ? c[i - 16384] : 0.0f;
    lap += (z < 127) ? c[i + 16384] : 0.0f;
    const float dt = dtp[0] * (1.0f / 50.0f);
    float v = cc + (Dv[i] * lap + Rv[i] * cc * (1.0f - cc)) * dt;
    cn[i] = fminf(fmaxf(v, 0.0f), 1.0f);
}

// ---------------------------------------------------------------------------
extern "C" void kernel_launch(void* const* d_in, const int* in_sizes, int n_in,
                              void* d_out, int out_size, void* d_ws, size_t ws_size,
                              hipStream_t stream) {
    const float* x    = (const float*)d_in[0];
    const float* dt   = (const float*)d_in[1];
    const float* w1   = (const float*)d_in[2];
    const float* b1   = (const float*)d_in[3];
    const float* g1g  = (const float*)d_in[4];
    const float* g1b  = (const float*)d_in[5];
    const float* w2   = (const float*)d_in[6];
    const float* b2   = (const float*)d_in[7];
    const float* g2g  = (const float*)d_in[8];
    const float* g2b  = (const float*)d_in[9];
    const float* w3   = (const float*)d_in[10];
    const float* b3   = (const float*)d_in[11];

    float* out  = (float*)d_out;
    float* outC = out;            // c_t1
    float* outD = out + VOL;      // D
    float* outR = out + 2 * VOL;  // rho

    // Workspace layout (floats). Peak use ~385 MB.
    float* ws     = (float*)d_ws;
    float* h1     = ws;                    // 16*VOL  (conv1 out, GN+SiLU in place)
    float* y2     = ws + 16 * VOL;         // 32*VOL  (conv2 out)
    float* wt1    = ws + 48 * VOL;         // 1728 (padded to 2048)
    float* wt2    = wt1 + 2048;            // 13824 (padded to 14336)
    float* stats  = wt2 + 14336;           // stats1[8] + stats2[16]
    float* stats1 = stats;
    float* stats2 = stats + 8;
    // c ping-pong buffers reuse the h1 region (h1 dead after conv2).
    float* cb0 = h1;
    float* cb1 = h1 + VOL;

    // Prep: zero stats, transpose weights for WMMA A-fragment loads.
    zero_kernel<<<1, 64, 0, stream>>>(stats, 64);
    wprep_kernel<4, 16><<<(16 * 4 * 27 + 255) / 256, 256, 0, stream>>>(w1, wt1);
    wprep_kernel<16, 32><<<(32 * 16 * 27 + 255) / 256, 256, 0, stream>>>(w2, wt2);

    const dim3 cgrid(2, DIM, DIM);

    // Layer 1: conv -> stats -> GN+SiLU (in place)
    conv3_wmma_kernel<4, 16><<<cgrid, 128, 0, stream>>>(x, wt1, b1, h1);
    gn_stats_kernel<<<dim3(VOL / 8192, 16), 256, 0, stream>>>(h1, stats1);
    gn_silu_kernel<<<(16 * VOL) / 256, 256, 0, stream>>>(h1, stats1, g1g, g1b);

    // Layer 2: conv -> stats -> GN+SiLU fused with 1x1x1 head -> D, rho
    conv3_wmma_kernel<16, 32><<<cgrid, 128, 0, stream>>>(h1, wt2, b2, y2);
    gn_stats_kernel<<<dim3(VOL / 8192, 32), 256, 0, stream>>>(y2, stats2);
    gn2_head_kernel<<<VOL / 256, 256, 0, stream>>>(y2, stats2, g2g, g2b, w3, b3, outD, outR);

    // Reaction-diffusion: c0 = x[:,0], 50 steps, final step writes d_out.
    hipMemcpyAsync(cb0, x, VOL * sizeof(float), hipMemcpyDeviceToDevice, stream);
    for (int s = 0; s < 50; ++s) {
        const float* src = (s & 1) ? cb1 : cb0;
        float*       dst = (s == 49) ? outC : ((s & 1) ? cb0 : cb1);
        rd_step_kernel<<<VOL / 256, 256, 0, stream>>>(src, dst, outD, outR, dt);
    }
}